// CGLayer_24335284699378
// MI455X (gfx1250) — compile-verified
//
#include <hip/hip_runtime.h>
#include <hip/hip_bf16.h>
#include <stdint.h>

// ---------------------------------------------------------------------------
// CDNA5 (gfx1250). f32-exact GEMMs via V_WMMA_F32_16X16X4_F32.
// - point-major activation layouts -> contiguous b64/b128 vector memory ops
// - K permuted (features first, rel-xyz rows 128..130, zero pad col 131)
// - weights staged in LDS (dynamic, 66/132 KB; WGP has 320 KB) with K-stride
//   132 (== 4 mod 64 banks) -> conflict-free b64 LDS reads
// - software-pipelined K loop: batch A loads, prefetch next B chunk
// Workspace requirement: ~440 MB.
// ---------------------------------------------------------------------------

typedef __attribute__((ext_vector_type(2))) float v2f;
typedef __attribute__((ext_vector_type(8))) float v8f;

#define B_   4
#define N_   16384
#define M_   2048
#define C_   128
#define K0_  131        // 128 feature rows + 3 rel-xyz rows (permuted order)
#define KP_  132        // padded K stride (132 mod 64 == 4 -> conflict-free)
#define O0_  128
#define O1_  256
#define NS_  32
#define P_   (M_ * NS_) // 65536 columns per batch
#define R2_  9.0f
#define EPS_ 1e-5f

// stats buffer layout (floats)
#define SUM0   0
#define SQ0    64
#define MEAN0  128
#define INV0   192
#define SUM1   256
#define SQ1    288
#define MEAN1  320
#define INV1   352
#define SUMA   384
#define SQA    512
#define MEANA  640
#define INVA   768
#define SUMB   896
#define SQB    1152
#define MEANB  1408
#define INVB   1664
#define SCALEA 1920
#define SHIFTA 2048
#define SCALEB 2176
#define SHIFTB 2432
#define STATS_FLOATS 4096

// ---------------------------------------------------------------------------
__global__ void k_zero_stats(float* stats) {
  int gid = blockIdx.x * blockDim.x + threadIdx.x;
  if (gid < STATS_FLOATS) stats[gid] = 0.0f;
}

// mean / inv-std (+ optional fused BN scale/shift) from accumulated sums
__global__ void k_finalize(float* stats, int sumOff, int sqOff, int meanOff,
                           int invOff, int scaleOff, int shiftOff,
                           const float* g, const float* bta,
                           int n, float count) {
  int i = threadIdx.x;
  if (i < n) {
    float mu  = stats[sumOff + i] / count;
    float var = stats[sqOff + i] / count - mu * mu;
    float inv = rsqrtf(var + EPS_);
    stats[meanOff + i] = mu;
    stats[invOff + i]  = inv;
    if (scaleOff >= 0) {
      float sc = inv * g[i];
      stats[scaleOff + i] = sc;
      stats[shiftOff + i] = bta[i] - mu * sc;
    }
  }
}

// pad + K-permute mlp_w0 (128x131) -> wp0 (128x132):
//   col 0..127   = w0[:, 3+col]    (feature rows)
//   col 128..130 = w0[:, col-128]  (rel-xyz rows)
//   col 131      = 0               (pad)
__global__ void k_prepw(const float* __restrict__ w0, float* __restrict__ wp0) {
  int gid = blockIdx.x * 256 + threadIdx.x;
  if (gid >= O0_ * KP_) return;
  int row = gid / KP_, col = gid - row * KP_;
  float v = 0.0f;
  if (col < 128) v = w0[row * K0_ + 3 + col];
  else if (col < 131) v = w0[row * K0_ + (col - 128)];
  wp0[gid] = v;
}

// LDS-tiled transpose: feat [B][C][N] -> featT [B][N][C]
__global__ void k_transpose(const float* __restrict__ in, float* __restrict__ out) {
  __shared__ float tile[32][33];
  int b = blockIdx.z;
  int n0 = blockIdx.x * 32, c0 = blockIdx.y * 32;
  int tx = threadIdx.x, ty = threadIdx.y;  // 32 x 8
#pragma unroll
  for (int j = 0; j < 32; j += 8)
    tile[ty + j][tx] = in[((size_t)(b * C_ + c0 + ty + j)) * N_ + n0 + tx];
  __syncthreads();
#pragma unroll
  for (int j = 0; j < 32; j += 8)
    out[((size_t)(b * N_ + n0 + ty + j)) * C_ + c0 + tx] = tile[tx][ty + j];
}

// ---------------------------------------------------------------------------
// shift MLP layer 0: t0[b][o][m] = w0[o][:] . ffps[b][m][:]  (o<64), + stats
__global__ void k_shift0(const float* __restrict__ ffps,
                         const float* __restrict__ w0,
                         float* __restrict__ t0, float* __restrict__ stats) {
  __shared__ float ssum[64], ssq[64];
  int tid = threadIdx.x;
  if (tid < 64) { ssum[tid] = 0.0f; ssq[tid] = 0.0f; }
  __syncthreads();
  int gid = blockIdx.x * 256 + tid;     // 8192 points
  int b = gid >> 11, m = gid & (M_ - 1);
  const float* x = ffps + (size_t)gid * 3;
  float x0 = x[0], x1 = x[1], x2 = x[2];
  for (int o = 0; o < 64; ++o) {
    float v = w0[o * 3] * x0 + w0[o * 3 + 1] * x1 + w0[o * 3 + 2] * x2;
    t0[((size_t)(b * 64 + o)) * M_ + m] = v;
    atomicAdd(&ssum[o], v);
    atomicAdd(&ssq[o], v * v);
  }
  __syncthreads();
  if (tid < 64) {
    atomicAdd(&stats[SUM0 + tid], ssum[tid]);
    atomicAdd(&stats[SQ0 + tid], ssq[tid]);
  }
}

// shift layer 1: bn0+relu on t0, then 64->3, + stats
__global__ void k_shift1(const float* __restrict__ t0,
                         const float* __restrict__ w1,
                         const float* __restrict__ g0,
                         const float* __restrict__ b0,
                         float* __restrict__ t1, float* __restrict__ stats) {
  __shared__ float ssum[3], ssq[3];
  int tid = threadIdx.x;
  if (tid < 3) { ssum[tid] = 0.0f; ssq[tid] = 0.0f; }
  __syncthreads();
  int gid = blockIdx.x * 256 + tid;
  int b = gid >> 11, m = gid & (M_ - 1);
  float a0 = 0.0f, a1 = 0.0f, a2 = 0.0f;
  for (int o = 0; o < 64; ++o) {
    float v = t0[((size_t)(b * 64 + o)) * M_ + m];
    float h = (v - stats[MEAN0 + o]) * stats[INV0 + o] * g0[o] + b0[o];
    h = fmaxf(h, 0.0f);
    a0 += w1[o] * h;
    a1 += w1[64 + o] * h;
    a2 += w1[128 + o] * h;
  }
  t1[((size_t)(b * 3 + 0)) * M_ + m] = a0;
  t1[((size_t)(b * 3 + 1)) * M_ + m] = a1;
  t1[((size_t)(b * 3 + 2)) * M_ + m] = a2;
  atomicAdd(&ssum[0], a0); atomicAdd(&ssq[0], a0 * a0);
  atomicAdd(&ssum[1], a1); atomicAdd(&ssq[1], a1 * a1);
  atomicAdd(&ssum[2], a2); atomicAdd(&ssq[2], a2 * a2);
  __syncthreads();
  if (tid < 3) {
    atomicAdd(&stats[SUM1 + tid], ssum[tid]);
    atomicAdd(&stats[SQ1 + tid], ssq[tid]);
  }
}

__global__ void k_shift_apply(const float* __restrict__ t1,
                              const float* __restrict__ g1,
                              const float* __restrict__ b1,
                              const float* __restrict__ stats,
                              float* __restrict__ nxyz) {
  int gid = blockIdx.x * 256 + threadIdx.x;
  int b = gid >> 11, m = gid & (M_ - 1);
  for (int j = 0; j < 3; ++j) {
    float v = t1[((size_t)(b * 3 + j)) * M_ + m];
    float h = (v - stats[MEAN1 + j]) * stats[INV1 + j] * g1[j] + b1[j];
    nxyz[(size_t)gid * 3 + j] = fmaxf(h, 0.0f);
  }
}

// ---------------------------------------------------------------------------
// Ball query: one wave per query point; ordered first-32 selection via
// ballot + prefix popcount; early exit; pad with first hit (0 if none).
__global__ void k_ballq(const float* __restrict__ bxyz,
                        const float* __restrict__ nxyz,
                        int* __restrict__ idxb) {
  int lane = threadIdx.x & 31;
  int q = blockIdx.x * 8 + (threadIdx.x >> 5);   // 8192 queries
  int b = q >> 11;
  const float* nx = nxyz + (size_t)q * 3;
  float qx = nx[0], qy = nx[1], qz = nx[2];
  const float* px = bxyz + (size_t)b * N_ * 3;
  int* dst = idxb + (size_t)q * NS_;
  int cnt = 0, first = 0;
  for (int base = 0; base < N_; base += 32) {
    int n = base + lane;
    float dx = px[n * 3] - qx;
    float dy = px[n * 3 + 1] - qy;
    float dz = px[n * 3 + 2] - qz;
    float d2 = dx * dx + dy * dy + dz * dz;
    bool hit = d2 < R2_;
    unsigned mask = (unsigned)__ballot(hit);
    if (cnt == 0 && mask) first = base + __ffs(mask) - 1;
    int pre = __popc(mask & ((1u << lane) - 1u));
    int pos = cnt + pre;
    if (hit && pos < NS_) dst[pos] = n;
    cnt += __popc(mask);
    if (cnt >= NS_) break;
  }
  for (int j = cnt + lane; j < NS_; j += 32) dst[j] = first;  // pad
}

// ---------------------------------------------------------------------------
// conv0: 131 -> 128, f32 WMMA 16x16x4. One wave = 16-point column tile,
// 8 output-channel tiles in registers. Weights staged in LDS (66 KB dynamic,
// stride 132 -> conflict-free). Software-pipelined: next-chunk B prefetched
// (global b64), per-chunk A tiles batched (ds_load_b64) before the WMMAs.
__global__ void k_mlp0(const float* __restrict__ featT,  // [B][N][128]
                       const float* __restrict__ bxyz,
                       const float* __restrict__ wp0,    // [128][132] padded
                       const float* __restrict__ nxyz,
                       const int* __restrict__ idxb,
                       float* __restrict__ c0T) {        // [B][P][128]
  extern __shared__ float lw[];                          // O0_ * KP_ floats
  for (int i = threadIdx.x; i < O0_ * KP_; i += 256) lw[i] = wp0[i];
  __syncthreads();

  int wave = threadIdx.x >> 5, lane = threadIdx.x & 31;
  int col = lane & 15, half = lane >> 4;
  int b = blockIdx.x >> 9;           // 512 blocks per batch
  int tile = blockIdx.x & 511;
  int p = tile * 128 + wave * 16 + col;
  int m = p >> 5, ks = p & 31;
  int gi = idxb[((size_t)b * M_ + m) * NS_ + ks];
  const float* nx = nxyz + ((size_t)b * M_ + m) * 3;
  const float* gx = bxyz + ((size_t)b * N_ + gi) * 3;
  float r0 = gx[0] - nx[0], r1 = gx[1] - nx[1], r2 = gx[2] - nx[2];
  const float* fp = featT + ((size_t)b * N_ + gi) * C_;  // this point's feats
  const float* la = lw + col * KP_ + 2 * half;           // lane's A base

  v8f zero = {0, 0, 0, 0, 0, 0, 0, 0};
  v8f acc[8];
#pragma unroll
  for (int i = 0; i < 8; ++i) acc[i] = zero;

  v2f tailb = half ? (v2f){r2, 0.0f} : (v2f){r0, r1};
  v2f bv = *(const v2f*)(fp + 2 * half);                 // chunk 0
  for (int t = 0; t < 32; ++t) {
    // prefetch next chunk's B operand (gathered feats, or rel-xyz tail)
    v2f bn = (t < 31) ? *(const v2f*)(fp + (t + 1) * 4 + 2 * half) : tailb;
    int k = t * 4;
    v2f a[8];
#pragma unroll
    for (int ot = 0; ot < 8; ++ot) a[ot] = *(const v2f*)(la + ot * 16 * KP_ + k);
#pragma unroll
    for (int ot = 0; ot < 8; ++ot)
      acc[ot] = __builtin_amdgcn_wmma_f32_16x16x4_f32(
          false, a[ot], false, bv, (short)0, acc[ot], false, false);
    bv = bn;
  }
  {  // tail chunk: rel-xyz rows 128..130 + zero pad
    v2f a[8];
#pragma unroll
    for (int ot = 0; ot < 8; ++ot) a[ot] = *(const v2f*)(la + ot * 16 * KP_ + 128);
#pragma unroll
    for (int ot = 0; ot < 8; ++ot)
      acc[ot] = __builtin_amdgcn_wmma_f32_16x16x4_f32(
          false, a[ot], false, bv, (short)0, acc[ot], false, false);
  }
  float* dst = c0T + (size_t)(b * (size_t)P_ + p) * O0_;
#pragma unroll
  for (int ot = 0; ot < 8; ++ot)
    *(v8f*)(dst + ot * 16 + 8 * half) = acc[ot];   // contiguous 32B stores
}

// conv1: 128 -> 256, BN(scale/shift)+relu fused on contiguous b64 loads.
// w1 staged in LDS (132 KB dynamic, re-strided 128 -> 132 for bank-freedom).
__global__ void k_mlp1(const float* __restrict__ c0T,   // [B][P][128]
                       const float* __restrict__ w1,    // [256][128]
                       const float* __restrict__ stats,
                       float* __restrict__ c1T) {       // [B][P][256]
  extern __shared__ float lw[];                         // O1_ * KP_ floats
  for (int i = threadIdx.x; i < O1_ * O0_; i += 256) {
    int row = i >> 7, kcol = i & 127;
    lw[row * KP_ + kcol] = w1[i];
  }
  __syncthreads();

  int wave = threadIdx.x >> 5, lane = threadIdx.x & 31;
  int col = lane & 15, half = lane >> 4;
  int b = blockIdx.x >> 9;
  int tile = blockIdx.x & 511;
  int p = tile * 128 + wave * 16 + col;
  const float* src = c0T + (size_t)(b * (size_t)P_ + p) * O0_;
  const float* la = lw + col * KP_ + 2 * half;

  v8f zero = {0, 0, 0, 0, 0, 0, 0, 0};
  v8f acc[16];
#pragma unroll
  for (int i = 0; i < 16; ++i) acc[i] = zero;

  v2f xv = *(const v2f*)(src + 2 * half);               // chunk 0 (raw)
  for (int t = 0; t < 32; ++t) {                        // K = 128
    v2f xn = (t < 31) ? *(const v2f*)(src + (t + 1) * 4 + 2 * half) : xv;
    int k = t * 4 + 2 * half;
    v2f sc = *(const v2f*)(stats + SCALEA + k);
    v2f sh = *(const v2f*)(stats + SHIFTA + k);
    v2f bv = { fmaxf(xv[0] * sc[0] + sh[0], 0.0f),
               fmaxf(xv[1] * sc[1] + sh[1], 0.0f) };
    v2f a[16];
#pragma unroll
    for (int ot = 0; ot < 16; ++ot)
      a[ot] = *(const v2f*)(la + ot * 16 * KP_ + t * 4);
#pragma unroll
    for (int ot = 0; ot < 16; ++ot)
      acc[ot] = __builtin_amdgcn_wmma_f32_16x16x4_f32(
          false, a[ot], false, bv, (short)0, acc[ot], false, false);
    xv = xn;
  }
  float* dst = c1T + (size_t)(b * (size_t)P_ + p) * O1_;
#pragma unroll
  for (int ot = 0; ot < 16; ++ot)
    *(v8f*)(dst + ot * 16 + 8 * half) = acc[ot];
}

// ---------------------------------------------------------------------------
// per-channel sum/sumsq over point-major pre-BN output [B*P][nch]
// thread = channel -> fully coalesced; blockDim.x == nch
__global__ void k_cstats(const float* __restrict__ x, float* __restrict__ stats,
                         int nch, int sumOff, int sqOff) {
  int ch = threadIdx.x;
  size_t rows = (size_t)B_ * P_;             // 262144
  size_t row0 = (size_t)blockIdx.x * (rows / 256);
  size_t row1 = row0 + rows / 256;
  float s = 0.0f, ss = 0.0f;
  for (size_t r = row0; r < row1; ++r) {
    float v = x[r * nch + ch];
    s += v; ss += v * v;
  }
  atomicAdd(&stats[sumOff + ch], s);
  atomicAdd(&stats[sqOff + ch], ss);
}

// BN + relu + max over 32 samples -> out[b][m][o]  (coalesced in o)
__global__ void k_out(const float* __restrict__ c1T,   // [B][P][256]
                      const float* __restrict__ stats,
                      float* __restrict__ out) {
  size_t gid = (size_t)blockIdx.x * 256 + threadIdx.x;  // 2,097,152 total
  int o = (int)(gid & 255);
  int m = (int)((gid >> 8) & (M_ - 1));
  int b = (int)(gid >> 19);
  float sc = stats[SCALEB + o], sh = stats[SHIFTB + o];
  const float* src = c1T + ((size_t)b * P_ + (size_t)m * NS_) * O1_ + o;
  float best = 0.0f;   // relu(max(x)) == max over relu'd samples
  for (int k = 0; k < NS_; ++k)
    best = fmaxf(best, src[(size_t)k * O1_] * sc + sh);
  out[gid] = best;
}

// ---------------------------------------------------------------------------
extern "C" void kernel_launch(void* const* d_in, const int* in_sizes, int n_in,
                              void* d_out, int out_size, void* d_ws, size_t ws_size,
                              hipStream_t stream) {
  const float* ffps = (const float*)d_in[0];
  const float* bxyz = (const float*)d_in[1];
  const float* feat = (const float*)d_in[2];
  const float* sw0  = (const float*)d_in[3];
  const float* sg0  = (const float*)d_in[4];
  const float* sb0  = (const float*)d_in[5];
  const float* sw1  = (const float*)d_in[6];
  const float* sg1  = (const float*)d_in[7];
  const float* sb1  = (const float*)d_in[8];
  const float* mw0  = (const float*)d_in[9];
  const float* mg0  = (const float*)d_in[10];
  const float* mb0  = (const float*)d_in[11];
  const float* mw1  = (const float*)d_in[12];
  const float* mg1  = (const float*)d_in[13];
  const float* mb1  = (const float*)d_in[14];
  float* out = (float*)d_out;

  char* ws = (char*)d_ws;
  size_t off = 0;
  float* stats = (float*)(ws + off); off += STATS_FLOATS * 4;
  float* wp0   = (float*)(ws + off); off += (size_t)O0_ * KP_ * 4;      // 68 KB
  float* t0    = (float*)(ws + off); off += (size_t)B_ * 64 * M_ * 4;   // 2 MB
  float* t1    = (float*)(ws + off); off += (size_t)B_ * 3 * M_ * 4;
  float* nxyz  = (float*)(ws + off); off += (size_t)B_ * M_ * 3 * 4;
  int*   idxb  = (int*)  (ws + off); off += (size_t)B_ * M_ * NS_ * 4;  // 1 MB
  float* featT = (float*)(ws + off); off += (size_t)B_ * N_ * C_ * 4;   // 33.5 MB
  float* c0T   = (float*)(ws + off); off += (size_t)B_ * O0_ * P_ * 4;  // 134 MB
  float* c1T   = (float*)(ws + off);                                    // 268 MB

  k_zero_stats<<<16, 256, 0, stream>>>(stats);
  k_prepw<<<(O0_ * KP_ + 255) / 256, 256, 0, stream>>>(mw0, wp0);
  k_transpose<<<dim3(N_ / 32, C_ / 32, B_), dim3(32, 8), 0, stream>>>(feat, featT);

  // shift MLP (3 -> 64 -> 3) with global BN stats
  k_shift0<<<32, 256, 0, stream>>>(ffps, sw0, t0, stats);
  k_finalize<<<1, 256, 0, stream>>>(stats, SUM0, SQ0, MEAN0, INV0, -1, -1,
                                    nullptr, nullptr, 64, 8192.0f);
  k_shift1<<<32, 256, 0, stream>>>(t0, sw1, sg0, sb0, t1, stats);
  k_finalize<<<1, 256, 0, stream>>>(stats, SUM1, SQ1, MEAN1, INV1, -1, -1,
                                    nullptr, nullptr, 3, 8192.0f);
  k_shift_apply<<<32, 256, 0, stream>>>(t1, sg1, sb1, stats, nxyz);

  // ball query (first 32 indices within radius, ascending)
  k_ballq<<<1024, 256, 0, stream>>>(bxyz, nxyz, idxb);

  // conv0 (131 -> 128) WMMA (LDS-staged weights), then BN stats
  k_mlp0<<<2048, 256, (size_t)O0_ * KP_ * 4, stream>>>(featT, bxyz, wp0, nxyz,
                                                       idxb, c0T);
  k_cstats<<<256, O0_, 0, stream>>>(c0T, stats, O0_, SUMA, SQA);
  k_finalize<<<1, 256, 0, stream>>>(stats, SUMA, SQA, MEANA, INVA, SCALEA, SHIFTA,
                                    mg0, mb0, 128, 262144.0f);

  // conv1 (128 -> 256) WMMA (LDS-staged weights, BN0+relu fused on load)
  k_mlp1<<<2048, 256, (size_t)O1_ * KP_ * 4, stream>>>(c0T, mw1, stats, c1T);
  k_cstats<<<256, O1_, 0, stream>>>(c1T, stats, O1_, SUMB, SQB);
  k_finalize<<<1, 256, 0, stream>>>(stats, SUMB, SQB, MEANB, INVB, SCALEB, SHIFTB,
                                    mg1, mb1, 256, 262144.0f);

  // BN + relu + max-pool over samples + transpose
  k_out<<<8192, 256, 0, stream>>>(c1T, stats, out);
}